// AttentionNet_75402445849004
// MI455X (gfx1250) — compile-verified
//
#include <hip/hip_runtime.h>
#include <hip/hip_bf16.h>

typedef float v2f __attribute__((ext_vector_type(2)));
typedef float v8f __attribute__((ext_vector_type(8)));

#define NF   128
#define NFIL 64
#define EF   16
#define EFIL 32
#define CLS  16

// ---------------------------------------------------------------------------
// init: acc1 (N*64) <- 0 ; d_out (N*16) <- 0 ; deg (N) <- 1.0 (self loops)
// ---------------------------------------------------------------------------
__global__ void k_init(float* __restrict__ acc1, float* __restrict__ out,
                       float* __restrict__ deg, int n) {
    long long i = (long long)blockIdx.x * blockDim.x + threadIdx.x;
    long long n64 = (long long)n * 64;
    if (i < n64) acc1[i] = 0.0f;
    if (i < (long long)n * CLS) out[i] = 0.0f;
    if (i < n) deg[i] = 1.0f;
}

// ---------------------------------------------------------------------------
// edge MLP: ew[e] = sigmoid(relu(edge_x[e] @ W1 + b1) @ W2 + b2)
// ---------------------------------------------------------------------------
__global__ void k_edge_mlp(const float* __restrict__ ex,
                           const float* __restrict__ W1, const float* __restrict__ b1,
                           const float* __restrict__ W2, const float* __restrict__ b2,
                           float* __restrict__ ew, int E) {
    __shared__ float sW1[EF * EFIL];
    __shared__ float sB1[EFIL];
    __shared__ float sW2[EFIL];
    __shared__ float sB2;
    int t = threadIdx.x;
    for (int i = t; i < EF * EFIL; i += 256) sW1[i] = W1[i];
    if (t < EFIL) { sB1[t] = b1[t]; sW2[t] = W2[t]; }
    if (t == 0) sB2 = b2[0];
    __syncthreads();

    int e = blockIdx.x * 256 + t;
    if (e >= E) return;

    const float4* exv = (const float4*)(ex + (size_t)e * EF);
    float4 q0 = exv[0], q1 = exv[1], q2 = exv[2], q3 = exv[3];
    float xr[EF] = {q0.x, q0.y, q0.z, q0.w, q1.x, q1.y, q1.z, q1.w,
                    q2.x, q2.y, q2.z, q2.w, q3.x, q3.y, q3.z, q3.w};

    float s = 0.0f;
#pragma unroll
    for (int j = 0; j < EFIL; ++j) {
        float h = sB1[j];
#pragma unroll
        for (int k = 0; k < EF; ++k) h = fmaf(xr[k], sW1[k * EFIL + j], h);
        h = fmaxf(h, 0.0f);
        s = fmaf(h, sW2[j], s);
    }
    s += sB2;
    ew[e] = 1.0f / (1.0f + __expf(-s));
}

// ---------------------------------------------------------------------------
// weighted degree: deg[tgt[e]] += ew[e]
// ---------------------------------------------------------------------------
__global__ void k_degree(const int* __restrict__ tgt, const float* __restrict__ ew,
                         float* __restrict__ deg, int E) {
    int e = blockIdx.x * blockDim.x + threadIdx.x;
    if (e >= E) return;
    atomicAdd(&deg[tgt[e]], ew[e]);
}

// dinv = deg > 0 ? rsqrt(deg) : 0
__global__ void k_dinv(const float* __restrict__ deg, float* __restrict__ dinv, int n) {
    int i = blockIdx.x * blockDim.x + threadIdx.x;
    if (i >= n) return;
    float d = deg[i];
    dinv[i] = (d > 0.0f) ? rsqrtf(d) : 0.0f;
}

// ---------------------------------------------------------------------------
// GEMM1: Y[N,64] = X[N,128] @ W[128,64]   (fp32 WMMA 16x16x4)
// block = 128 threads (4 waves); block handles 16 rows; wave w -> cols [16w,16w+16)
// ---------------------------------------------------------------------------
__global__ void k_gemm1(const float* __restrict__ X, const float* __restrict__ W,
                        float* __restrict__ Y, int nrows) {
    __shared__ float sW[NF * NFIL];      // 32 KB
    __shared__ float sX[16 * 129];       // padded pitch vs LDS bank conflicts
    int t = threadIdx.x;
    for (int i = t; i < NF * NFIL; i += 128) sW[i] = W[i];

    int rowBase = blockIdx.x * 16;
    for (int i = t; i < 16 * NF; i += 128) {
        int m = i >> 7, k = i & 127;
        int r = rowBase + m; if (r >= nrows) r = nrows - 1;   // clamp (safe dup)
        sX[m * 129 + k] = X[(size_t)r * NF + k];
    }
    __syncthreads();

    int wave = t >> 5, lane = t & 31;
    int n = lane & 15, kh = lane >> 4, m = lane & 15;
    int col0 = wave * 16;

    v8f acc = {};
#pragma unroll
    for (int k = 0; k < NF; k += 4) {
        v2f a, b;
        a.x = sX[m * 129 + k + 2 * kh];
        a.y = sX[m * 129 + k + 2 * kh + 1];
        b.x = sW[(k + 2 * kh) * NFIL + col0 + n];
        b.y = sW[(k + 2 * kh + 1) * NFIL + col0 + n];
        acc = __builtin_amdgcn_wmma_f32_16x16x4_f32(false, a, false, b,
                                                    (short)0, acc, false, false);
    }
#pragma unroll
    for (int v = 0; v < 8; ++v) {
        int row = rowBase + v + 8 * kh;
        if (row < nrows) Y[(size_t)row * NFIL + col0 + n] = acc[v];
    }
}

// ---------------------------------------------------------------------------
// GEMM2: Y[N,16] = H[N,64] @ W[64,16]   (fp32 WMMA 16x16x4)
// block = 128 threads (4 waves); wave handles one 16-row tile
// ---------------------------------------------------------------------------
__global__ void k_gemm2(const float* __restrict__ H, const float* __restrict__ W,
                        float* __restrict__ Y, int nrows, int nTiles) {
    __shared__ float sW[NFIL * CLS];     // 4 KB
    int t = threadIdx.x;
    for (int i = t; i < NFIL * CLS; i += 128) sW[i] = W[i];
    __syncthreads();

    int wave = t >> 5, lane = t & 31;
    int tile = blockIdx.x * 4 + wave;
    if (tile >= nTiles) return;          // wave-uniform branch: EXEC stays full

    int n = lane & 15, kh = lane >> 4, m = lane & 15;
    int rowBase = tile * 16;
    int arow = rowBase + m; if (arow >= nrows) arow = nrows - 1;
    const float* __restrict__ A = H + (size_t)arow * NFIL;

    v8f acc = {};
#pragma unroll
    for (int k = 0; k < NFIL; k += 4) {
        v2f a, b;
        a.x = A[k + 2 * kh];
        a.y = A[k + 2 * kh + 1];
        b.x = sW[(k + 2 * kh) * CLS + n];
        b.y = sW[(k + 2 * kh + 1) * CLS + n];
        acc = __builtin_amdgcn_wmma_f32_16x16x4_f32(false, a, false, b,
                                                    (short)0, acc, false, false);
    }
#pragma unroll
    for (int v = 0; v < 8; ++v) {
        int row = rowBase + v + 8 * kh;
        if (row < nrows) Y[(size_t)row * CLS + n] = acc[v];
    }
}

// ---------------------------------------------------------------------------
// scatter layer 1: acc1[tgt,f] += dinv[src]*ew*dinv[tgt] * xw[src,f]   (64 feats)
// ---------------------------------------------------------------------------
__global__ void k_scatter1(const int* __restrict__ src, const int* __restrict__ tgt,
                           const float* __restrict__ ew, const float* __restrict__ dinv,
                           const float* __restrict__ xw, float* __restrict__ acc1,
                           long long E) {
    long long gid = (long long)blockIdx.x * blockDim.x + threadIdx.x;
    long long e = gid >> 6;
    int f = (int)(gid & 63);
    if (e >= E) return;
    int s = src[e], d = tgt[e];
    float nrm = dinv[s] * ew[e] * dinv[d];
    atomicAdd(&acc1[(size_t)d * NFIL + f], nrm * xw[(size_t)s * NFIL + f]);
}

// self-loop fold + bias + relu (in place on acc1)
__global__ void k_bias_relu(float* __restrict__ acc1, const float* __restrict__ xw,
                            const float* __restrict__ dinv, const float* __restrict__ bc1,
                            int n) {
    long long i = (long long)blockIdx.x * blockDim.x + threadIdx.x;
    if (i >= (long long)n * NFIL) return;
    int node = (int)(i >> 6), f = (int)(i & 63);
    float di = dinv[node];
    float v = acc1[i] + di * di * xw[i] + bc1[f];
    acc1[i] = fmaxf(v, 0.0f);
}

// ---------------------------------------------------------------------------
// scatter layer 2: out[tgt,f] += norm * h1w[src,f]   (16 feats)
// ---------------------------------------------------------------------------
__global__ void k_scatter2(const int* __restrict__ src, const int* __restrict__ tgt,
                           const float* __restrict__ ew, const float* __restrict__ dinv,
                           const float* __restrict__ h1w, float* __restrict__ out,
                           long long E) {
    long long gid = (long long)blockIdx.x * blockDim.x + threadIdx.x;
    long long e = gid >> 4;
    int f = (int)(gid & 15);
    if (e >= E) return;
    int s = src[e], d = tgt[e];
    float nrm = dinv[s] * ew[e] * dinv[d];
    atomicAdd(&out[(size_t)d * CLS + f], nrm * h1w[(size_t)s * CLS + f]);
}

// self-loop + bias + log_softmax (in place on out)
__global__ void k_final(float* __restrict__ out, const float* __restrict__ h1w,
                        const float* __restrict__ dinv, const float* __restrict__ bc2,
                        int n) {
    int i = blockIdx.x * blockDim.x + threadIdx.x;
    if (i >= n) return;
    float di = dinv[i];
    float sl = di * di;
    float z[CLS];
    float mx = -INFINITY;
#pragma unroll
    for (int f = 0; f < CLS; ++f) {
        float v = out[(size_t)i * CLS + f] + sl * h1w[(size_t)i * CLS + f] + bc2[f];
        z[f] = v;
        mx = fmaxf(mx, v);
    }
    float s = 0.0f;
#pragma unroll
    for (int f = 0; f < CLS; ++f) s += __expf(z[f] - mx);
    float lse = mx + __logf(s);
#pragma unroll
    for (int f = 0; f < CLS; ++f) out[(size_t)i * CLS + f] = z[f] - lse;
}

// ---------------------------------------------------------------------------
extern "C" void kernel_launch(void* const* d_in, const int* in_sizes, int n_in,
                              void* d_out, int out_size, void* d_ws, size_t ws_size,
                              hipStream_t stream) {
    const float* x    = (const float*)d_in[0];
    const int*   ei   = (const int*)  d_in[1];
    const float* ex   = (const float*)d_in[2];
    const float* W1   = (const float*)d_in[3];
    const float* b1   = (const float*)d_in[4];
    const float* W2   = (const float*)d_in[5];
    const float* b2   = (const float*)d_in[6];
    const float* Wc1  = (const float*)d_in[7];
    const float* bc1  = (const float*)d_in[8];
    const float* Wc2  = (const float*)d_in[9];
    const float* bc2  = (const float*)d_in[10];
    float* out = (float*)d_out;

    const int N = in_sizes[0] / NF;      // 100000
    const int E = in_sizes[1] / 2;       // 1600000
    const int* src = ei;
    const int* tgt = ei + E;

    // workspace layout (floats)
    float* ws   = (float*)d_ws;
    float* ew   = ws;                         // E
    float* deg  = ew  + E;                    // N
    float* dinv = deg + N;                    // N
    float* xw   = dinv + N;                   // 64*N
    float* acc1 = xw  + (size_t)NFIL * N;     // 64*N  (becomes h1 in place)
    float* h1w  = acc1 + (size_t)NFIL * N;    // 16*N

    const int BT = 256;
    long long n64 = (long long)N * NFIL;

    k_init<<<(unsigned)((n64 + BT - 1) / BT), BT, 0, stream>>>(acc1, out, deg, N);
    k_edge_mlp<<<(E + BT - 1) / BT, BT, 0, stream>>>(ex, W1, b1, W2, b2, ew, E);
    k_degree<<<(E + BT - 1) / BT, BT, 0, stream>>>(tgt, ew, deg, E);
    k_dinv<<<(N + BT - 1) / BT, BT, 0, stream>>>(deg, dinv, N);

    int rowTiles = (N + 15) / 16;
    k_gemm1<<<rowTiles, 128, 0, stream>>>(x, Wc1, xw, N);

    long long s1 = (long long)E * NFIL;
    k_scatter1<<<(unsigned)((s1 + BT - 1) / BT), BT, 0, stream>>>(src, tgt, ew, dinv,
                                                                  xw, acc1, E);
    k_bias_relu<<<(unsigned)((n64 + BT - 1) / BT), BT, 0, stream>>>(acc1, xw, dinv,
                                                                    bc1, N);
    k_gemm2<<<(rowTiles + 3) / 4, 128, 0, stream>>>(acc1, Wc2, h1w, N, rowTiles);

    long long s2 = (long long)E * CLS;
    k_scatter2<<<(unsigned)((s2 + BT - 1) / BT), BT, 0, stream>>>(src, tgt, ew, dinv,
                                                                  h1w, out, E);
    k_final<<<(N + BT - 1) / BT, BT, 0, stream>>>(out, h1w, dinv, bc2, N);
}